// MultiHeadAttention_9268539425425
// MI455X (gfx1250) — compile-verified
//
#include <hip/hip_runtime.h>
#include <hip/hip_bf16.h>

// ---------------------------------------------------------------------------
// Physics-biased multi-head attention for MI455X (gfx1250), fp32 WMMA path.
// B=2, S=2048, D_MODEL=64, H=12, D_K=5 (padded to 8 for WMMA K-chunks).
// Memory-bound: bias read (403MB, NT load) + attn write (403MB, NT store)
// dominate. Scores for a 16-query tile stay resident in LDS (320KB/WGP on
// CDNA5) so each of those tensors touches HBM exactly once.
// ---------------------------------------------------------------------------

typedef __attribute__((ext_vector_type(2))) float v2f;
typedef __attribute__((ext_vector_type(8))) float v8f;

#define BB 2
#define SS 2048
#define DM 64
#define NH 12
#define DK 5
#define DI 60          // NH*DK
#define PROW 2049      // padded LDS row stride (2049 % 64 == 1 -> conflict-free columns)
#define KT_TILES (SS / 16)   // 128 key tiles per row

static __device__ __forceinline__ v8f wmma_f32_4(v2f a, v2f b, v8f c) {
    // V_WMMA_F32_16X16X4_F32 : D = A(16x4) * B(4x16) + C(16x16)
    return __builtin_amdgcn_wmma_f32_16x16x4_f32(false, a, false, b, (short)0, c,
                                                 false, false);
}

// ---------------------------------------------------------------------------
// Kernel 1: QKV projections.  y = X[4096,64] @ W[64,60] + b  -> [4096,60]
// grid (256, 3), block 128 (4 waves; wave w owns output columns [16w,16w+16)).
// Out-of-width lanes (col>=60) load a clamped in-bounds address; the garbage
// they contribute only lands in C columns those same lanes own, which are
// never stored -> no divergent exec-masking in the inner loop.
// ---------------------------------------------------------------------------
__global__ void proj_kernel(const float* __restrict__ Xq,
                            const float* __restrict__ Xk,
                            const float* __restrict__ Xv,
                            const float* __restrict__ Wq, const float* __restrict__ bq,
                            const float* __restrict__ Wk, const float* __restrict__ bk,
                            const float* __restrict__ Wv, const float* __restrict__ bv,
                            float* __restrict__ qp, float* __restrict__ kp,
                            float* __restrict__ vp)
{
    const int which = blockIdx.y;
    const float* X = (which == 0) ? Xq : (which == 1) ? Xk : Xv;
    const float* W = (which == 0) ? Wq : (which == 1) ? Wk : Wv;
    const float* bs = (which == 0) ? bq : (which == 1) ? bk : bv;
    float* O = (which == 0) ? qp : (which == 1) ? kp : vp;

    const int mbase = blockIdx.x * 16;
    const int wave = threadIdx.x >> 5;
    const int lane = threadIdx.x & 31;
    const int half = lane >> 4;     // 0: lanes 0-15, 1: lanes 16-31
    const int m    = lane & 15;
    const int col  = wave * 16 + m;           // output column (N index)
    const int wcol = (col < DI) ? col : 0;    // clamped, branch-free address

    v8f c = {};
    const float* xrow = X + (size_t)(mbase + m) * DM;
    for (int kk = 0; kk < DM; kk += 4) {
        const int k0 = kk + half * 2;
        v2f a, b;
        a.x = xrow[k0];
        a.y = xrow[k0 + 1];
        b.x = W[(size_t)k0 * DI + wcol];
        b.y = W[(size_t)(k0 + 1) * DI + wcol];
        c = wmma_f32_4(a, b, c);
    }
    if (col < DI) {
        const float bias = bs[col];
        #pragma unroll
        for (int r = 0; r < 8; ++r) {
            const int row = mbase + r + half * 8;   // C layout: M = r + half*8
            O[(size_t)row * DI + col] = c[r] + bias;
        }
    }
}

// ---------------------------------------------------------------------------
// Kernel 2: attention core.  One block = one (b, h, 16-query tile).
// 256 threads = 8 waves; wave w owns key tiles {w, w+8, ...} of 128.
// Dynamic LDS: sP[16][PROW] scores + sRed[8][256] ctx reduction.
// ---------------------------------------------------------------------------
__global__ void attn_kernel(const float* __restrict__ qp,
                            const float* __restrict__ kp,
                            const float* __restrict__ vp,
                            const float* __restrict__ bias,
                            const float* __restrict__ lam_p,
                            float* __restrict__ attn_out,
                            float* __restrict__ ctx)
{
    extern __shared__ float smem[];
    float* sP   = smem;                 // [16][PROW]
    float* sRed = smem + 16 * PROW;     // [8][256]

    const int bid   = blockIdx.x;           // 0 .. B*H*(S/16)-1
    const int qt    = bid & (KT_TILES - 1); // query tile within row
    const int h     = (bid >> 7) % NH;
    const int b     = bid / (KT_TILES * NH);
    const int qbase = qt * 16;

    const int wave = threadIdx.x >> 5;
    const int lane = threadIdx.x & 31;
    const int half = lane >> 4;
    const int m    = lane & 15;

    const float lam   = lam_p[0];
    const float scale = 0.44721359549995793f;   // 1/sqrt(5)
    const int   d0    = half * 2;               // A/B K-offset for this half

    // ---- q A-fragments (16 x 8, d=5..7 zero-padded), shared by all key tiles
    const float* qrow = qp + ((size_t)b * SS + qbase + m) * DI + h * DK;
    v2f a0, a1;
    a0.x = qrow[d0];                       // d in {0,2}
    a0.y = qrow[d0 + 1];                   // d in {1,3}
    a1.x = (half == 0) ? qrow[4] : 0.0f;   // d=4 valid only in half 0
    a1.y = 0.0f;

    // ---- phase 1: scores = q.k^T * scale + lam*bias, into LDS
    const float* bbase =
        bias + (((size_t)(b * NH + h)) * SS + qbase) * SS;
    for (int kt = wave; kt < KT_TILES; kt += 8) {
        const int kbase = kt * 16;
        const float* krow = kp + ((size_t)b * SS + kbase + m) * DI + h * DK;
        v2f b0, b1;
        b0.x = krow[d0];
        b0.y = krow[d0 + 1];
        b1.x = (half == 0) ? krow[4] : 0.0f;
        b1.y = 0.0f;

        v8f c = {};
        c = wmma_f32_4(a0, b0, c);
        c = wmma_f32_4(a1, b1, c);

        if (kt + 8 < KT_TILES)   // speculative prefetch of next bias tile
            __builtin_prefetch(bbase + (size_t)(half * 8) * SS + kbase + 128 + m, 0, 1);

        #pragma unroll
        for (int r = 0; r < 8; ++r) {
            const int M = r + half * 8;
            // bias is a 403MB single-use stream: non-temporal load (TH=NT)
            const float bv = __builtin_nontemporal_load(
                bbase + (size_t)M * SS + kbase + m);
            sP[M * PROW + kbase + m] = c[r] * scale + lam * bv;
        }
    }
    __syncthreads();

    // ---- phase 2: softmax per row (wave w handles rows 2w, 2w+1), write attn
    #pragma unroll
    for (int rr = 0; rr < 2; ++rr) {
        const int row = wave * 2 + rr;
        float* pr = sP + row * PROW;

        float mx = -3.402823466e38f;
        for (int k = lane; k < SS; k += 32) mx = fmaxf(mx, pr[k]);
        #pragma unroll
        for (int off = 16; off > 0; off >>= 1)
            mx = fmaxf(mx, __shfl_xor(mx, off, 32));

        float sum = 0.0f;
        for (int k = lane; k < SS; k += 32) {
            const float e = __expf(pr[k] - mx);
            pr[k] = e;
            sum += e;
        }
        #pragma unroll
        for (int off = 16; off > 0; off >>= 1)
            sum += __shfl_xor(sum, off, 32);

        const float inv = 1.0f / sum;
        float* arow =
            attn_out + (((size_t)(b * NH + h)) * SS + qbase + row) * SS;
        for (int k = lane; k < SS; k += 32) {
            const float p = pr[k] * inv;
            pr[k] = p;
            // attn is a 403MB single-use stream: non-temporal store (TH=NT)
            __builtin_nontemporal_store(p, arow + k);
        }
    }
    __syncthreads();

    // ---- phase 3: ctx = P(16 x 2048) @ v(2048 x 5pad16) via WMMA, K=4 chunks.
    // Lanes with m>=5 load a clamped v address; their garbage lands only in
    // C columns N>=5, which are discarded at the guarded ctx store below.
    const int dd = (m < DK) ? m : 0;
    v8f cc = {};
    for (int kt = wave; kt < KT_TILES; kt += 8) {
        const int kbase = kt * 16;
        #pragma unroll
        for (int step = 0; step < 4; ++step) {
            const int k0 = kbase + step * 4 + d0;   // key index of B rows
            v2f a, bv2;
            a.x = sP[m * PROW + k0];                // conflict-free (stride 2049)
            a.y = sP[m * PROW + k0 + 1];
            const float* v0 = vp + ((size_t)b * SS + k0) * DI + h * DK;
            bv2.x = v0[dd];                         // B[K=key][N=d]
            bv2.y = v0[DI + dd];                    // key k0+1
            cc = wmma_f32_4(a, bv2, cc);
        }
    }
    #pragma unroll
    for (int r = 0; r < 8; ++r)
        sRed[wave * 256 + r * 32 + lane] = cc[r];
    __syncthreads();

    // cross-wave reduce: thread t sums element t over all 8 waves
    const int tid = threadIdx.x;
    float s = 0.0f;
    #pragma unroll
    for (int w = 0; w < 8; ++w) s += sRed[w * 256 + tid];
    const int r  = tid >> 5;
    const int l2 = tid & 31;
    const int M  = r + (l2 >> 4) * 8;
    const int N  = l2 & 15;
    if (N < DK)
        ctx[((size_t)b * SS + qbase + M) * DI + h * DK + N] = s;
}

// ---------------------------------------------------------------------------
// Kernel 3: out = residual + ctx[4096,60] @ Wo[60,64] + bo.
// grid 256, block 128 (4 waves; wave w owns output columns [16w,16w+16)).
// K=60 is exactly 15 WMMA steps of K=4 — no padding needed.
// ---------------------------------------------------------------------------
__global__ void outproj_kernel(const float* __restrict__ ctx,
                               const float* __restrict__ Wo,
                               const float* __restrict__ bo,
                               const float* __restrict__ Qin,
                               float* __restrict__ out)
{
    const int mbase = blockIdx.x * 16;
    const int wave = threadIdx.x >> 5;
    const int lane = threadIdx.x & 31;
    const int half = lane >> 4;
    const int m    = lane & 15;
    const int col  = wave * 16 + m;            // always < 64

    v8f c = {};
    const float* crow = ctx + (size_t)(mbase + m) * DI;
    for (int kk = 0; kk < DI; kk += 4) {
        const int k0 = kk + half * 2;
        v2f a, b;
        a.x = crow[k0];
        a.y = crow[k0 + 1];
        b.x = Wo[(size_t)k0 * DM + col];
        b.y = Wo[(size_t)(k0 + 1) * DM + col];
        c = wmma_f32_4(a, b, c);
    }
    const float bias = bo[col];
    #pragma unroll
    for (int r = 0; r < 8; ++r) {
        const int row = mbase + r + half * 8;
        out[(size_t)row * DM + col] =
            c[r] + bias + Qin[(size_t)row * DM + col];
    }
}

// ---------------------------------------------------------------------------
extern "C" void kernel_launch(void* const* d_in, const int* in_sizes, int n_in,
                              void* d_out, int out_size, void* d_ws, size_t ws_size,
                              hipStream_t stream) {
    const float* Q    = (const float*)d_in[0];
    const float* K    = (const float*)d_in[1];
    const float* V    = (const float*)d_in[2];
    const float* bias = (const float*)d_in[3];
    const float* lam  = (const float*)d_in[4];
    const float* Wq   = (const float*)d_in[5];
    const float* bq   = (const float*)d_in[6];
    const float* Wk   = (const float*)d_in[7];
    const float* bk   = (const float*)d_in[8];
    const float* Wv   = (const float*)d_in[9];
    const float* bv   = (const float*)d_in[10];
    const float* Wo   = (const float*)d_in[11];
    const float* bo   = (const float*)d_in[12];

    float* out  = (float*)d_out;                    // [B,S,64]
    float* attn = out + (size_t)BB * SS * DM;       // [B,H,S,S]

    const size_t NPROJ = (size_t)BB * SS * DI;      // 245760 floats
    float* qp  = (float*)d_ws;
    float* kp  = qp + NPROJ;
    float* vp  = kp + NPROJ;
    float* ctx = vp + NPROJ;

    // 1) QKV projections
    proj_kernel<<<dim3((BB * SS) / 16, 3), 128, 0, stream>>>(
        Q, K, V, Wq, bq, Wk, bk, Wv, bv, qp, kp, vp);

    // 2) attention core — needs > 64KB dynamic LDS (CDNA5 WGP has 320KB)
    const int smem_bytes = (16 * PROW + 8 * 256) * (int)sizeof(float); // 139,328 B
    hipFuncSetAttribute((const void*)attn_kernel,
                        hipFuncAttributeMaxDynamicSharedMemorySize, smem_bytes);
    attn_kernel<<<BB * NH * (SS / 16), 256, smem_bytes, stream>>>(
        qp, kp, vp, bias, lam, attn, ctx);

    // 3) output projection + residual
    outproj_kernel<<<(BB * SS) / 16, 128, 0, stream>>>(ctx, Wo, bo, Q, out);
}